// MaskBatchNorm2d_17076789969318
// MI455X (gfx1250) — compile-verified
//
#include <hip/hip_runtime.h>

#define EPS_F 1e-5f

typedef float v2f __attribute__((ext_vector_type(2)));
typedef float v4f __attribute__((ext_vector_type(4)));
typedef float v8f __attribute__((ext_vector_type(8)));
typedef unsigned char u8x4 __attribute__((ext_vector_type(4)));

// Full wave32 sum using V_WMMA_F32_16X16X4_F32.
// A (16x4 f32, 2 VGPRs/lane): VGPR0 -> A[m][0]=p[m] (lanes 0-15),
// A[m][2]=p[m+16] (lanes 16-31); VGPR1=0. B = ones(4x16)
// => D[m][n] = p[m] + p[m+16] replicated across n. Summing the 8 D VGPRs
// per lane gives half-sums in each 16-lane half; one xor-16 shuffle finishes.
// EXEC is all-ones at every call site (no divergence around calls).
__device__ __forceinline__ float wave_sum32(float v) {
  v2f a; a[0] = v;    a[1] = 0.0f;
  v2f b; b[0] = 1.0f; b[1] = 1.0f;
  v8f c = {};
  v8f d = __builtin_amdgcn_wmma_f32_16x16x4_f32(
      /*neg_a=*/false, a, /*neg_b=*/false, b,
      /*c_mod=*/(short)0, c, /*reuse_a=*/false, /*reuse_b=*/false);
  float s = ((d[0] + d[1]) + (d[2] + d[3])) + ((d[4] + d[5]) + (d[6] + d[7]));
  s += __shfl_xor(s, 16, 32);
  return s;
}

// One block per channel d (256 blocks x 512 threads = 16 waves/WGP).
// Streams 32 planes of 4096 f32 (16 KB): 1024 v4f / plane, 2 per thread.
// Regular (RT) loads on purpose: fills L2 so the apply pass re-reads x from L2.
__global__ __launch_bounds__(512) void mbn_stats_kernel(
    const float* __restrict__ x, const unsigned char* __restrict__ mask,
    const float* __restrict__ gamma, const float* __restrict__ beta,
    float* __restrict__ ws) {
  const int d   = blockIdx.x;   // channel
  const int tid = threadIdx.x;  // 0..511
  const int D = 256, HW = 4096, NB = 32;

  float s1 = 0.0f, s2 = 0.0f, cn = 0.0f;

  for (int bb = 0; bb < NB; ++bb) {
    const v4f*  xp = (const v4f*)(x + ((size_t)bb * D + d) * HW);
    const u8x4* mp = (const u8x4*)(mask + (size_t)bb * HW);
    if (bb + 1 < NB && tid < 256) {
      // 256 lanes x 64B = the whole next 16 KB plane -> global_prefetch_b8
      __builtin_prefetch((const char*)(x + ((size_t)(bb + 1) * D + d) * HW) +
                             (size_t)tid * 64, 0, 3);
    }
#pragma unroll
    for (int i = 0; i < 2; ++i) {
      const int idx = tid + i * 512;
      const v4f  xv = xp[idx];
      const u8x4 mv = mp[idx];
      const float m0 = mv[0] ? 0.0f : xv[0];
      const float m1 = mv[1] ? 0.0f : xv[1];
      const float m2 = mv[2] ? 0.0f : xv[2];
      const float m3 = mv[3] ? 0.0f : xv[3];
      s1 += (m0 + m1) + (m2 + m3);
      s2 = fmaf(m0, xv[0], fmaf(m1, xv[1], fmaf(m2, xv[2], fmaf(m3, xv[3], s2))));
      cn += ((mv[0] ? 0.0f : 1.0f) + (mv[1] ? 0.0f : 1.0f)) +
            ((mv[2] ? 0.0f : 1.0f) + (mv[3] ? 0.0f : 1.0f));
    }
  }

  // wave32 reduction via WMMA, then combine the 16 waves through LDS.
  s1 = wave_sum32(s1);
  s2 = wave_sum32(s2);
  cn = wave_sum32(cn);

  __shared__ float red[3][16];
  const int wave = tid >> 5, lane = tid & 31;
  if (lane == 0) { red[0][wave] = s1; red[1][wave] = s2; red[2][wave] = cn; }
  __syncthreads();
  if (tid == 0) {
    float S1 = 0.0f, S2 = 0.0f, C = 0.0f;
#pragma unroll
    for (int w = 0; w < 16; ++w) { S1 += red[0][w]; S2 += red[1][w]; C += red[2][w]; }
    const float mean = S1 / C;
    const float var  = S2 / C - mean * mean;
    const float inv  = gamma[d] * rsqrtf(var + EPS_F);
    ws[d]       = inv;                         // scale
    ws[256 + d] = beta[d] - mean * inv;        // bias: y = x*inv + bias
  }
}

// One block per (b,d) plane (8192 blocks x 256 threads): inv/bias are
// block-uniform scalars. x loads NT (last use, keep L2 for in-flight data),
// out stores NT (don't evict x from L2 while pass 2 is still reading it).
__global__ __launch_bounds__(256) void mbn_apply_kernel(
    const float* __restrict__ x, const unsigned char* __restrict__ mask,
    const float* __restrict__ ws, float* __restrict__ out) {
  const int D = 256, HW = 4096;
  const int plane = blockIdx.x;      // b*256 + d
  const int d  = plane & (D - 1);
  const int bb = plane >> 8;
  const float inv  = ws[d];
  const float bias = ws[256 + d];

  const v4f*  xp = (const v4f*)(x + (size_t)plane * HW);
  v4f*        op = (v4f*)(out + (size_t)plane * HW);
  const u8x4* mp = (const u8x4*)(mask + (size_t)bb * HW);
  const int tid = threadIdx.x;

#pragma unroll
  for (int i = 0; i < 4; ++i) {
    const int idx = tid + i * 256;
    const v4f  xv = __builtin_nontemporal_load(xp + idx);
    const u8x4 mv = mp[idx];
    v4f yv;
    yv[0] = mv[0] ? xv[0] : fmaf(xv[0], inv, bias);
    yv[1] = mv[1] ? xv[1] : fmaf(xv[1], inv, bias);
    yv[2] = mv[2] ? xv[2] : fmaf(xv[2], inv, bias);
    yv[3] = mv[3] ? xv[3] : fmaf(xv[3], inv, bias);
    __builtin_nontemporal_store(yv, op + idx);
  }
}

extern "C" void kernel_launch(void* const* d_in, const int* in_sizes, int n_in,
                              void* d_out, int out_size, void* d_ws, size_t ws_size,
                              hipStream_t stream) {
  (void)in_sizes; (void)n_in; (void)out_size; (void)ws_size;
  const float*         x     = (const float*)d_in[0];
  const unsigned char* mask  = (const unsigned char*)d_in[1];  // jnp.bool_ -> 1 byte/elem
  const float*         gamma = (const float*)d_in[2];
  const float*         beta  = (const float*)d_in[3];
  float*               out   = (float*)d_out;
  float*               ws    = (float*)d_ws;  // [0..255]=inv, [256..511]=bias

  mbn_stats_kernel<<<256, 512, 0, stream>>>(x, mask, gamma, beta, ws);
  mbn_apply_kernel<<<8192, 256, 0, stream>>>(x, mask, ws, out);
}